// ConvSelfAttention_78597901517129
// MI455X (gfx1250) — compile-verified
//
#include <hip/hip_runtime.h>

// ---------------------------------------------------------------------------
// ConvSelfAttention for MI455X (gfx1250), full-fp32 path via V_WMMA_F32_16X16X4_F32
// Shapes: BS=8, C=256, N=64*64=4096, DK=64.
// Workspace layout (floats):
//   qT  [BS][N][DK]   offset 0        (8 MB)  -- Q stored transposed (n-major)
//   K   [BS][DK][N]   offset 2097152  (8 MB)
//   V   [BS][DK][N]   offset 4194304  (8 MB)
//   L   [BS][N]       offset 6291456  (128 KB)  L_i = rowmax_i + ln(rowsum_i)
// Total ~25.3 MB of d_ws.
// ---------------------------------------------------------------------------

static constexpr int kBS = 8;
static constexpr int kC  = 256;   // channels
static constexpr int kN  = 4096;  // spatial (64*64)
static constexpr int kDK = 64;    // head dim

typedef __attribute__((ext_vector_type(2))) float v2f;
typedef __attribute__((ext_vector_type(8))) float v8f;

__device__ __forceinline__ v8f wmma_f32(v2f a, v2f b, v8f c) {
  // D(16x16) = A(16x4) * B(4x16) + C, fp32 in/out
  return __builtin_amdgcn_wmma_f32_16x16x4_f32(false, a, false, b,
                                               (short)0, c, false, false);
}

// ---------------------------------------------------------------------------
// Kernel 1: qkv = Wqkv(192x256) @ x(256x4096) per batch.
// One wave computes a 16(M) x 64(N) tile. Q rows (<64) are stored transposed.
// ---------------------------------------------------------------------------
__global__ void qkv_kernel(const float* __restrict__ x,
                           const float* __restrict__ Wqkv,
                           float* __restrict__ qT,
                           float* __restrict__ Kd,
                           float* __restrict__ Vd) {
  const int L    = threadIdx.x;
  const int col  = L & 15;
  const int half = L >> 4;

  int blk = blockIdx.x;                 // BS * 12 * 64 blocks
  const int b   = blk / (12 * 64);
  const int rem = blk % (12 * 64);
  const int m0  = (rem / 64) * 16;
  const int n0  = (rem % 64) * 64;

  const float* xb   = x + (size_t)b * kC * kN;
  const float* Arow = Wqkv + (size_t)(m0 + col) * kC + 2 * half;

  v8f c0 = {}, c1 = {}, c2 = {}, c3 = {};
  const int n = n0 + col;

#pragma unroll 4
  for (int kk = 0; kk < kC; kk += 4) {
    v2f a = *(const v2f*)(Arow + kk);                  // A: rows m0+col, K=kk+2h..+1
    const float* xr0 = xb + (size_t)(kk + 2 * half) * kN;
    const float* xr1 = xr0 + kN;
    v2f b0 = { xr0[n],      xr1[n]      };
    v2f b1 = { xr0[n + 16], xr1[n + 16] };
    v2f b2 = { xr0[n + 32], xr1[n + 32] };
    v2f b3 = { xr0[n + 48], xr1[n + 48] };
    c0 = wmma_f32(a, b0, c0);
    c1 = wmma_f32(a, b1, c1);
    c2 = wmma_f32(a, b2, c2);
    c3 = wmma_f32(a, b3, c3);
  }

  v8f cs[4] = { c0, c1, c2, c3 };
#pragma unroll
  for (int t = 0; t < 4; ++t) {
#pragma unroll
    for (int r = 0; r < 8; ++r) {
      const int row = m0 + r + 8 * half;
      const int nc  = n0 + 16 * t + col;
      const float vv = cs[t][r];
      if (row < 64) {                 // Q -> transposed
        qT[((size_t)b * kN + nc) * kDK + row] = vv;
      } else if (row < 128) {         // K
        Kd[((size_t)b * kDK + (row - 64)) * kN + nc] = vv;
      } else {                        // V
        Vd[((size_t)b * kDK + (row - 128)) * kN + nc] = vv;
      }
    }
  }
}

// ---------------------------------------------------------------------------
// Kernel 2: per (batch, i-tile of 16) compute L_i = max_j s[i,j] + ln sum_j exp(s-max)
// s[i,j] = sum_d q[d,i]*k[d,j]; online per-lane softmax stats, shfl reduce.
// ---------------------------------------------------------------------------
__global__ void stats_kernel(const float* __restrict__ qT,
                             const float* __restrict__ Kd,
                             float* __restrict__ Lrow) {
  const int L    = threadIdx.x;
  const int col  = L & 15;
  const int half = L >> 4;
  const int b  = blockIdx.x >> 8;
  const int i0 = (blockIdx.x & 255) * 16;

  // Preload A operands (qT rows i0..i0+15, all 64 d) : 16 x v2f
  v2f aq[16];
  const float* qrow = qT + ((size_t)b * kN + i0 + col) * kDK + 2 * half;
#pragma unroll
  for (int s = 0; s < 16; ++s) aq[s] = *(const v2f*)(qrow + 4 * s);

  const float* Kb = Kd + (size_t)b * kDK * kN;

  float mr[8], zr[8];
#pragma unroll
  for (int r = 0; r < 8; ++r) { mr[r] = -3.0e38f; zr[r] = 0.0f; }

  for (int jt = 0; jt < 256; ++jt) {
    const int n = jt * 16 + col;
    v8f c = {};
#pragma unroll
    for (int s = 0; s < 16; ++s) {
      const float* kr = Kb + (size_t)(4 * s + 2 * half) * kN;
      v2f bb = { kr[n], kr[n + kN] };
      c = wmma_f32(aq[s], bb, c);
    }
#pragma unroll
    for (int r = 0; r < 8; ++r) {
      const float v  = c[r];
      const float mn = fmaxf(mr[r], v);
      zr[r] = zr[r] * __expf(mr[r] - mn) + __expf(v - mn);
      mr[r] = mn;
    }
  }

  // Reduce across the 16 lanes of each half (each lane held one column slice)
#pragma unroll
  for (int r = 0; r < 8; ++r) {
    float m = mr[r], z = zr[r];
#pragma unroll
    for (int mask = 1; mask < 16; mask <<= 1) {
      const float om = __shfl_xor(m, mask, 16);
      const float oz = __shfl_xor(z, mask, 16);
      const float mn = fmaxf(m, om);
      z = z * __expf(m - mn) + oz * __expf(om - mn);
      m = mn;
    }
    if (col == 0)
      Lrow[(size_t)b * kN + i0 + 8 * half + r] = m + __logf(z);
  }
}

// ---------------------------------------------------------------------------
// Kernel 3: per (batch, j-tile) : Y[64,16] = sum_i V[:,i] * exp(s[i,j]-L_i),
// then fused O[256,16] = Wout @ Y * gamma + x residual.
// K-tile B-operands cached in registers across all 256 i-tiles.
// ---------------------------------------------------------------------------
__global__ void attn_out_kernel(const float* __restrict__ qT,
                                const float* __restrict__ Kd,
                                const float* __restrict__ Vd,
                                const float* __restrict__ Lrow,
                                const float* __restrict__ Wout,
                                const float* __restrict__ x,
                                const float* __restrict__ gamma,
                                float* __restrict__ out) {
  __shared__ float Pt[16 * 16];   // P' tile, row-major [i_local][j_local]
  __shared__ float ys[64 * 16];   // Y tile,  row-major [d][j_local]

  const int L    = threadIdx.x;
  const int col  = L & 15;
  const int half = L >> 4;
  const int b  = blockIdx.x >> 8;
  const int j0 = (blockIdx.x & 255) * 16;
  const int n  = j0 + col;

  const float* Kb = Kd + (size_t)b * kDK * kN;
  const float* Vb = Vd + (size_t)b * kDK * kN;
  const float* Lb = Lrow + (size_t)b * kN;

  // Cache K j-tile as B operands for the whole i loop (32 VGPRs)
  v2f kb2[16];
#pragma unroll
  for (int s = 0; s < 16; ++s) {
    const float* kr = Kb + (size_t)(4 * s + 2 * half) * kN;
    kb2[s] = { kr[n], kr[n + kN] };
  }

  v8f yacc[4] = { v8f{}, v8f{}, v8f{}, v8f{} };

  for (int it = 0; it < 256; ++it) {
    const int i0 = it * 16;

    // S tile = Q^T(i,d) @ K(d,j)
    const float* qrow = qT + ((size_t)b * kN + i0 + col) * kDK + 2 * half;
    v8f c = {};
#pragma unroll
    for (int s = 0; s < 16; ++s) {
      v2f a = *(const v2f*)(qrow + 4 * s);
      c = wmma_f32(a, kb2[s], c);
    }

    // P' = exp(S - L_row), written to LDS in row-major for B-operand reload
    const float* lp = Lb + i0 + 8 * half;
    const float4 la  = *(const float4*)(lp);
    const float4 lb4 = *(const float4*)(lp + 4);
    const float lv[8] = { la.x, la.y, la.z, la.w, lb4.x, lb4.y, lb4.z, lb4.w };
#pragma unroll
    for (int r = 0; r < 8; ++r)
      Pt[(r + 8 * half) * 16 + col] = __expf(c[r] - lv[r]);
    __syncthreads();

    // Y += V(d, i-chunk) @ P'(i-chunk, j)
#pragma unroll
    for (int ci = 0; ci < 4; ++ci) {
      v2f bp = { Pt[(4 * ci + 2 * half) * 16 + col],
                 Pt[(4 * ci + 2 * half + 1) * 16 + col] };
      const float* vbase = Vb + (size_t)col * kN + (i0 + 4 * ci + 2 * half);
#pragma unroll
      for (int dt = 0; dt < 4; ++dt) {
        v2f a = *(const v2f*)(vbase + (size_t)(16 * dt) * kN);
        yacc[dt] = wmma_f32(a, bp, yacc[dt]);
      }
    }
    __syncthreads();
  }

  // Spill Y to LDS for the output-projection B operands
#pragma unroll
  for (int dt = 0; dt < 4; ++dt)
#pragma unroll
    for (int r = 0; r < 8; ++r)
      ys[(16 * dt + r + 8 * half) * 16 + col] = yacc[dt][r];
  __syncthreads();

  // O tile = Wout(256x64) @ Y(64x16) * gamma + x
  const float g = gamma[0];
  const float* xb = x + (size_t)b * kC * kN;
  float* ob = out + (size_t)b * kC * kN;

  for (int mt = 0; mt < 16; ++mt) {
    const int m0 = mt * 16;
    const float* wr = Wout + (size_t)(m0 + col) * kDK + 2 * half;
    v8f c = {};
#pragma unroll
    for (int s = 0; s < 16; ++s) {
      v2f a  = *(const v2f*)(wr + 4 * s);
      v2f bb = { ys[(4 * s + 2 * half) * 16 + col],
                 ys[(4 * s + 2 * half + 1) * 16 + col] };
      c = wmma_f32(a, bb, c);
    }
#pragma unroll
    for (int r = 0; r < 8; ++r) {
      const int row = m0 + r + 8 * half;
      const size_t idx = (size_t)row * kN + n;
      ob[idx] = c[r] * g + xb[idx];
    }
  }
}

// ---------------------------------------------------------------------------
extern "C" void kernel_launch(void* const* d_in, const int* in_sizes, int n_in,
                              void* d_out, int out_size, void* d_ws, size_t ws_size,
                              hipStream_t stream) {
  const float* x     = (const float*)d_in[0];  // (8,256,64,64) fp32
  const float* Wqkv  = (const float*)d_in[1];  // (192,256)
  const float* Wout  = (const float*)d_in[2];  // (256,64)
  const float* gamma = (const float*)d_in[3];  // (1,)
  float* out = (float*)d_out;                  // (8,256,64,64) fp32

  float* ws = (float*)d_ws;                    // needs ~25.3 MB
  float* qT = ws;                              // [8][4096][64]
  float* Kd = ws + 2097152;                    // [8][64][4096]
  float* Vd = ws + 4194304;                    // [8][64][4096]
  float* Lr = ws + 6291456;                    // [8][4096]

  qkv_kernel<<<kBS * 12 * 64, 32, 0, stream>>>(x, Wqkv, qT, Kd, Vd);
  stats_kernel<<<kBS * 256, 32, 0, stream>>>(qT, Kd, Lr);
  attn_out_kernel<<<kBS * 256, 32, 0, stream>>>(qT, Kd, Vd, Lr, Wout, x, gamma, out);
  (void)in_sizes; (void)n_in; (void)out_size; (void)ws_size;
}